// NeuralCollapseLoss_29918742184188
// MI455X (gfx1250) — compile-verified
//
#include <hip/hip_runtime.h>

typedef __attribute__((ext_vector_type(2))) float v2f;
typedef __attribute__((ext_vector_type(8))) float v8f;

#define NC_EPSILON 5.0f
#define NC_D 256

// One wave (32 lanes) handles 16 samples.
// diff fragments are arranged in the CDNA5 16x4 f32 A-matrix layout:
//   lanes 0-15  : row r = lane,    K = k0+0, k0+1
//   lanes 16-31 : row r = lane-16, K = k0+2, k0+3
// B layout mirrors A with N = lane, so wmma(d, d, acc) accumulates
// diff * diff^T; after the K loop the diagonal holds dist^2 per sample.
__global__ __launch_bounds__(128) void nc_loss_main_kernel(
    const float* __restrict__ features,
    const float* __restrict__ means,
    const long long* __restrict__ labels,
    float* __restrict__ partials,
    int numTiles)
{
    __shared__ float waveSums[4];

    const int tid  = threadIdx.x;
    const int wave = tid >> 5;
    const int lane = tid & 31;
    const int tile = blockIdx.x * 4 + wave;   // wave-uniform

    float contrib = 0.0f;

    if (tile < numTiles) {                    // wave-uniform branch: EXEC stays all-1s
        const int r    = lane & 15;
        const int half = lane >> 4;           // 0 or 1
        const int s    = tile * 16 + r;
        const int cls  = (int)labels[s];      // labels are int64, classes < 1000

        const float* frow = features + (size_t)s   * NC_D;
        const float* mrow = means    + (size_t)cls * NC_D;

        __builtin_prefetch(frow, 0, 0);       // global_prefetch_b8
        __builtin_prefetch(mrow, 0, 0);

        v8f acc = {};
        const int kbase = half * 2;

        #pragma unroll 4
        for (int k0 = 0; k0 < NC_D; k0 += 4) {
            const int k = k0 + kbase;         // 8B-aligned (k even)
            v2f f = *(const v2f*)(frow + k);  // global_load_b64
            v2f m = *(const v2f*)(mrow + k);  // global_load_b64 (L2-resident)
            v2f d;
            d.x = f.x - m.x;
            d.y = f.y - m.y;
            // D = A*B + C with A == B == diff fragment -> Gram matrix accumulate
            acc = __builtin_amdgcn_wmma_f32_16x16x4_f32(
                false, d, false, d, (short)0, acc, false, false);
        }

        // Diagonal extraction from the 16x16 f32 C/D layout:
        //   lanes 0-7  : C[v,lane]      diagonal at v == lane
        //   lanes 24-31: C[v+8,lane-16] diagonal at v == lane-24
        float d2 = 0.0f;
        #pragma unroll
        for (int v = 0; v < 8; ++v) {
            bool take = (lane < 16) ? (v == lane) : (v == lane - 24);
            float cv = acc[v];
            d2 = take ? cv : d2;
        }
        const bool valid = (lane < 8) || (lane >= 24);
        const float dist = __builtin_sqrtf(d2);
        const float loss = fmaxf(NC_EPSILON - dist, 0.0f);
        contrib = valid ? loss : 0.0f;
    }

    // wave32 butterfly reduction
    #pragma unroll
    for (int off = 16; off >= 1; off >>= 1)
        contrib += __shfl_xor(contrib, off, 32);

    if (lane == 0) waveSums[wave] = contrib;
    __syncthreads();
    if (tid == 0) {
        // fixed-order sum -> deterministic
        partials[blockIdx.x] = waveSums[0] + waveSums[1] + waveSums[2] + waveSums[3];
    }
}

__global__ __launch_bounds__(256) void nc_loss_reduce_kernel(
    const float* __restrict__ partials, int n,
    float* __restrict__ out, float invB)
{
    __shared__ float sm[256];
    float s = 0.0f;
    for (int i = threadIdx.x; i < n; i += 256) s += partials[i];  // fixed order per thread
    sm[threadIdx.x] = s;
    __syncthreads();
    #pragma unroll
    for (int stride = 128; stride >= 1; stride >>= 1) {
        if ((int)threadIdx.x < stride) sm[threadIdx.x] += sm[threadIdx.x + stride];
        __syncthreads();
    }
    if (threadIdx.x == 0) out[0] = sm[0] * invB;
}

extern "C" void kernel_launch(void* const* d_in, const int* in_sizes, int n_in,
                              void* d_out, int out_size, void* d_ws, size_t ws_size,
                              hipStream_t stream) {
    const float*     features = (const float*)d_in[0];      // [B, 256] f32
    const float*     means    = (const float*)d_in[1];      // [1000, 256] f32
    const long long* labels   = (const long long*)d_in[2];  // [B] int64

    const int B        = in_sizes[2];          // 262144
    const int numTiles = (B + 15) / 16;        // 16 samples per wave
    const int blocks   = (numTiles + 3) / 4;   // 4 waves (128 threads) per block

    float* partials = (float*)d_ws;            // blocks * 4 bytes (~16 KB)

    nc_loss_main_kernel<<<blocks, 128, 0, stream>>>(
        features, means, labels, partials, numTiles);
    nc_loss_reduce_kernel<<<1, 256, 0, stream>>>(
        partials, blocks, (float*)d_out, 1.0f / (float)B);
}